// Loss_fairness_regularization_6579889897516
// MI455X (gfx1250) — compile-verified
//
#include <hip/hip_runtime.h>
#include <hip/hip_bf16.h>
#include <math.h>

// ---------------------------------------------------------------------------
// Fairness-regularized loss: 9 global sums over N=16.7M rows, 3 scalar outputs.
// Memory-bound: 256 MiB streamed once -> ~11.5us floor @ 23.3 TB/s.
// Strategy: b128 non-temporal loads, grid-stride wave32 accumulation,
//           cross-lane reduce via V_WMMA_F32_16X16X4_F32 (A=acc, B=ones),
//           deterministic two-kernel tree reduction (no atomics).
// ---------------------------------------------------------------------------

typedef __attribute__((ext_vector_type(2))) float v2f;
typedef __attribute__((ext_vector_type(4))) float v4f;
typedef __attribute__((ext_vector_type(8))) float v8f;

#define NBLK 1024
#define NTHR 256
#define NWAVE (NTHR / 32)
#define NACC 9

// Exact 32-lane sum using the f32 WMMA unit.
// A (16x4): V0 = x, V1 = 0  => A[m][0]=x[m], A[m][2]=x[m+16], others 0.
// B (4x16) = ones           => D[m][n] = x[m] + x[m+16] (rowsum, all n).
// D layout: lanes (n, n+16) over the 8 D VGPRs hold a full column, so
// sum of d[0..7] + shfl_xor(16) = total over all 32 lanes. EXEC must be ~0.
__device__ __forceinline__ float wave_sum32(float x) {
  v2f a; a.x = x;    a.y = 0.0f;
  v2f b; b.x = 1.0f; b.y = 1.0f;
  v8f c = {0.f, 0.f, 0.f, 0.f, 0.f, 0.f, 0.f, 0.f};
  v8f d = __builtin_amdgcn_wmma_f32_16x16x4_f32(
      /*neg_a=*/false, a, /*neg_b=*/false, b,
      /*c_mod=*/(short)0, c, /*reuse_a=*/false, /*reuse_b=*/false);
  float s = (d[0] + d[1]) + (d[2] + d[3]);
  s += (d[4] + d[5]) + (d[6] + d[7]);
  s += __shfl_xor(s, 16, 32);   // combine rows 0..7 half with rows 8..15 half
  return s;
}

__global__ __launch_bounds__(NTHR) void fair_partial_kernel(
    const float* __restrict__ pred,   // (N,1) f32
    const float* __restrict__ lt,     // (N,3) f32: [y, married, single]
    float* __restrict__ part,         // (NBLK, 9) f32 block partials
    int n4)                           // N/4 groups of 4 rows
{
  float acc[NACC];
#pragma unroll
  for (int i = 0; i < NACC; ++i) acc[i] = 0.0f;

  const int tid    = blockIdx.x * NTHR + threadIdx.x;
  const int stride = NBLK * NTHR;

  for (int g = tid; g < n4; g += stride) {
    const int i0 = g * 4;
    // 4 rows of pred: one b128 NT load
    v4f p4 = __builtin_nontemporal_load((const v4f*)(pred + i0));
    // 4 rows x 3 cols of label_true: three b128 NT loads (48B, 16B-aligned)
    const v4f* tb = (const v4f*)(lt + 3 * i0);
    v4f t0 = __builtin_nontemporal_load(tb + 0);
    v4f t1 = __builtin_nontemporal_load(tb + 1);
    v4f t2 = __builtin_nontemporal_load(tb + 2);

    const float pp[4]  = {p4.x, p4.y, p4.z, p4.w};
    const float tt[12] = {t0.x, t0.y, t0.z, t0.w,
                          t1.x, t1.y, t1.z, t1.w,
                          t2.x, t2.y, t2.z, t2.w};
#pragma unroll
    for (int j = 0; j < 4; ++j) {
      const float p = pp[j];
      const float y = tt[3 * j + 0];
      const float m = tt[3 * j + 1];
      const float s = tt[3 * j + 2];

      // clip is a numerical no-op for this data (p in [1e-6, 1-1e-6]) but keep it
      const float pc  = fminf(fmaxf(p, 1e-12f), 1.0f - 1e-12f);
      // 1-pc is exact (Sterbenz) for pc>=0.5, well-conditioned below.
      const float lp  = __logf(pc);
      const float l1p = __logf(1.0f - pc);
      acc[0] += -(y * lp + (1.0f - y) * l1p);

      const float pb   = (p >= 0.5f) ? 1.0f : 0.0f;
      const float pby  = pb * y;        // exact 0/1 products
      const float pbny = pb - pby;
      acc[1] += m * pby;    // tp_married
      acc[2] += m * pbny;   // fp_married
      acc[3] += m * y;      // pos_married  (tp+fn)
      acc[4] += m;          // cnt_married
      acc[5] += s * pby;    // tp_single
      acc[6] += s * pbny;   // fp_single
      acc[7] += s * y;      // pos_single
      acc[8] += s;          // cnt_single
    }
  }

  // Wave-level reduce (WMMA) -> LDS -> block partial. EXEC is full here.
  __shared__ float red[NWAVE][NACC];
  const int wave = threadIdx.x >> 5;
  const int lane = threadIdx.x & 31;
#pragma unroll
  for (int i = 0; i < NACC; ++i) {
    const float t = wave_sum32(acc[i]);
    if (lane == 0) red[wave][i] = t;
  }
  __syncthreads();
  if (threadIdx.x < NACC) {
    float t = 0.0f;
#pragma unroll
    for (int w = 0; w < NWAVE; ++w) t += red[w][threadIdx.x];
    part[blockIdx.x * NACC + threadIdx.x] = t;
  }
}

__global__ __launch_bounds__(NTHR) void fair_final_kernel(
    const float* __restrict__ part,   // (NBLK, 9)
    float* __restrict__ out,          // [loss_fair, ce, eo]
    float invN)
{
  float acc[NACC];
#pragma unroll
  for (int i = 0; i < NACC; ++i) acc[i] = 0.0f;

  for (int r = threadIdx.x; r < NBLK; r += NTHR) {
    const float* row = part + r * NACC;
#pragma unroll
    for (int i = 0; i < NACC; ++i) acc[i] += row[i];
  }

  __shared__ float red[NWAVE][NACC];
  const int wave = threadIdx.x >> 5;
  const int lane = threadIdx.x & 31;
#pragma unroll
  for (int i = 0; i < NACC; ++i) {
    const float t = wave_sum32(acc[i]);
    if (lane == 0) red[wave][i] = t;
  }
  __syncthreads();

  if (threadIdx.x == 0) {
    float tot[NACC];
#pragma unroll
    for (int i = 0; i < NACC; ++i) {
      float t = 0.0f;
#pragma unroll
      for (int w = 0; w < NWAVE; ++w) t += red[w][i];
      tot[i] = t;
    }
    const float ce    = tot[0] * invN;
    const float tp_m  = tot[1], fp_m = tot[2], pos_m = tot[3], cnt_m = tot[4];
    const float tp_s  = tot[5], fp_s = tot[6], pos_s = tot[7], cnt_s = tot[8];
    // tp+fn = pos ; fp+tn = cnt-pos
    const float tpr_m = tp_m / fmaxf(pos_m, 1.0f);
    const float tpr_s = tp_s / fmaxf(pos_s, 1.0f);
    const float fpr_m = fp_m / fmaxf(cnt_m - pos_m, 1.0f);
    const float fpr_s = fp_s / fmaxf(cnt_s - pos_s, 1.0f);
    const float eo    = fabsf(tpr_m - tpr_s) + fabsf(fpr_m - fpr_s);
    out[0] = 0.5f * ce + 0.5f * eo;   // loss_fair (beta = 0.5)
    out[1] = ce;
    out[2] = eo;
  }
}

extern "C" void kernel_launch(void* const* d_in, const int* in_sizes, int n_in,
                              void* d_out, int out_size, void* d_ws, size_t ws_size,
                              hipStream_t stream) {
  const float* pred = (const float*)d_in[0];   // (N,1) float32
  const float* lt   = (const float*)d_in[1];   // (N,3) float32
  float* out  = (float*)d_out;                 // 3 floats
  float* part = (float*)d_ws;                  // NBLK*9 floats = 36 KB scratch

  const int n  = in_sizes[0];   // N (pred is N*1 flat)
  const int n4 = n / 4;         // N is a multiple of 4

  fair_partial_kernel<<<NBLK, NTHR, 0, stream>>>(pred, lt, part, n4);
  fair_final_kernel<<<1, NTHR, 0, stream>>>(part, out, 1.0f / (float)n);
}